// Placer_82832739271226
// MI455X (gfx1250) — compile-verified
//
#include <hip/hip_runtime.h>
#include <stdint.h>

#define NUM_CELLS 1000000
#define NUM_NETS  1500000
#define NUM_PINS  6000000
#define GAMMA_F   4.0f
#define INV_GAMMA 0.25f

#define TILE  2048
#define BLOCK 256
#define CHUNK (TILE / BLOCK)   // 8 consecutive (sorted-net) pins per lane

typedef unsigned int v4u __attribute__((ext_vector_type(4)));
typedef int          v4i __attribute__((ext_vector_type(4)));
typedef int          v8i __attribute__((ext_vector_type(8)));

// ---------- order-preserving float <-> uint key (atomicMax on floats incl. negatives)
__device__ __forceinline__ unsigned f2key(float f) {
  unsigned u = __float_as_uint(f);
  return (u & 0x80000000u) ? ~u : (u | 0x80000000u);
}
__device__ __forceinline__ float key2f(unsigned k) {
  return (k & 0x80000000u) ? __uint_as_float(k & 0x7FFFFFFFu)
                           : __uint_as_float(~k);
}

__device__ __forceinline__ unsigned lds_byte_off(const void* p) {
  // generic pointer to LDS: low 32 bits are the LDS byte offset (aperture mapping)
  return (unsigned)(unsigned long long)p;
}

// ---------- Tensor Data Mover: 1-D contiguous tile Global -> LDS (D# per CDNA5 ISA)
__device__ __forceinline__ void tdm_load_1d(unsigned lds_off, const void* gptr,
                                            unsigned n_elems, unsigned tile_elems,
                                            unsigned data_size_code /*2=4B,3=8B*/) {
  unsigned long long ga = (unsigned long long)gptr;
  v4u g0;
  g0.x = 1u;                                                  // count=1, no gather
  g0.y = lds_off;                                             // lds_addr (bytes)
  g0.z = (unsigned)(ga & 0xFFFFFFFFull);                      // global_addr[31:0]
  g0.w = ((unsigned)(ga >> 32) & 0x01FFFFFFu) | 0x80000000u;  // addr[56:32] | type=2
  v8i g1;
  g1[0] = (int)(data_size_code << 16);                        // workgroup_mask=0, data_size
  g1[1] = (int)((n_elems & 0xFFFFu) << 16);                   // tensor_dim0[15:0]
  g1[2] = (int)((n_elems >> 16) | (1u << 16));                // tensor_dim0[31:16], tensor_dim1=1
  g1[3] = (int)(tile_elems << 16);                            // tile_dim0
  g1[4] = 0;                                                  // tile_dim1=0 (1-D), tile_dim2=0
  g1[5] = (int)n_elems;                                       // tensor_dim0_stride lo32
  g1[6] = 0;                                                  // stride hi, dim1_stride lo
  g1[7] = 0;
  v4i z4 = {0, 0, 0, 0};
#if defined(__clang_major__) && (__clang_major__ >= 23)
  v8i z8 = {0, 0, 0, 0, 0, 0, 0, 0};
  __builtin_amdgcn_tensor_load_to_lds(g0, g1, z4, z4, z8, 0);
#else
  __builtin_amdgcn_tensor_load_to_lds(g0, g1, z4, z4, 0);
#endif
}

__device__ __forceinline__ void stage_tile(int tile0, int nrem,
                                           const int* pin_net, const int* pin_cell,
                                           const float* pin_offsets,
                                           int* s_net, int* s_cell, float* s_off) {
  if ((threadIdx.x >> 5) == 0) {   // one wave drives the DMA engine
    tdm_load_1d(lds_byte_off(s_net),  pin_net  + tile0, (unsigned)nrem, TILE, 2u);
    tdm_load_1d(lds_byte_off(s_cell), pin_cell + tile0, (unsigned)nrem, TILE, 2u);
    tdm_load_1d(lds_byte_off(s_off),  pin_offsets + (size_t)tile0 * 2,
                (unsigned)nrem, TILE, 3u);                    // 8B per pin (x,y)
    __builtin_amdgcn_s_wait_tensorcnt(0);
  }
  __syncthreads();
}

// ---------- pass 1: per-net running max of {x, -x, y, -y} via key atomicMax
__global__ void __launch_bounds__(BLOCK) pass_minmax(
    const float* __restrict__ cells_pos, const float* __restrict__ pin_offsets,
    const int* __restrict__ pin_cell, const int* __restrict__ pin_net,
    unsigned* __restrict__ keys) {
  __shared__ int   s_net[TILE];
  __shared__ int   s_cell[TILE];
  __shared__ float s_off[2 * TILE];

  const int tile0 = blockIdx.x * TILE;
  int nrem = NUM_PINS - tile0; if (nrem > TILE) nrem = TILE;
  stage_tile(tile0, nrem, pin_net, pin_cell, pin_offsets, s_net, s_cell, s_off);

  unsigned* kxp = keys;
  unsigned* kxn = keys + NUM_NETS;
  unsigned* kyp = keys + 2 * NUM_NETS;
  unsigned* kyn = keys + 3 * NUM_NETS;

  const int lane0 = threadIdx.x * CHUNK;
  int   cur = -1;
  float mxp = 0.f, mxn = 0.f, myp = 0.f, myn = 0.f;
  for (int i = 0; i < CHUNK; ++i) {
    const int li = lane0 + i;
    if (tile0 + li >= NUM_PINS) break;
    const int net  = s_net[li];
    const int cell = s_cell[li];
    const float x = cells_pos[2 * cell + 0] + s_off[2 * li + 0];
    const float y = cells_pos[2 * cell + 1] + s_off[2 * li + 1];
    if (net != cur) {
      if (cur >= 0) {
        atomicMax(&kxp[cur], f2key(mxp)); atomicMax(&kxn[cur], f2key(mxn));
        atomicMax(&kyp[cur], f2key(myp)); atomicMax(&kyn[cur], f2key(myn));
      }
      cur = net; mxp = x; mxn = -x; myp = y; myn = -y;
    } else {
      mxp = fmaxf(mxp, x);  mxn = fmaxf(mxn, -x);
      myp = fmaxf(myp, y);  myn = fmaxf(myn, -y);
    }
  }
  if (cur >= 0) {
    atomicMax(&kxp[cur], f2key(mxp)); atomicMax(&kxn[cur], f2key(mxn));
    atomicMax(&kyp[cur], f2key(myp)); atomicMax(&kyn[cur], f2key(myn));
  }
}

// ---------- pass 2: per-net sum of exp((v - m)/gamma), run-aggregated atomicAdd
__global__ void __launch_bounds__(BLOCK) pass_sums(
    const float* __restrict__ cells_pos, const float* __restrict__ pin_offsets,
    const int* __restrict__ pin_cell, const int* __restrict__ pin_net,
    const unsigned* __restrict__ keys, float* __restrict__ sums) {
  __shared__ int   s_net[TILE];
  __shared__ int   s_cell[TILE];
  __shared__ float s_off[2 * TILE];

  const int tile0 = blockIdx.x * TILE;
  int nrem = NUM_PINS - tile0; if (nrem > TILE) nrem = TILE;
  stage_tile(tile0, nrem, pin_net, pin_cell, pin_offsets, s_net, s_cell, s_off);

  const unsigned* kxp = keys;
  const unsigned* kxn = keys + NUM_NETS;
  const unsigned* kyp = keys + 2 * NUM_NETS;
  const unsigned* kyn = keys + 3 * NUM_NETS;
  float* sxp = sums;
  float* sxn = sums + NUM_NETS;
  float* syp = sums + 2 * NUM_NETS;
  float* syn = sums + 3 * NUM_NETS;

  const int lane0 = threadIdx.x * CHUNK;
  int   cur = -1;
  float mxp = 0.f, mxn = 0.f, myp = 0.f, myn = 0.f;
  float axp = 0.f, axn = 0.f, ayp = 0.f, ayn = 0.f;
  for (int i = 0; i < CHUNK; ++i) {
    const int li = lane0 + i;
    if (tile0 + li >= NUM_PINS) break;
    const int net  = s_net[li];
    const int cell = s_cell[li];
    const float x = cells_pos[2 * cell + 0] + s_off[2 * li + 0];
    const float y = cells_pos[2 * cell + 1] + s_off[2 * li + 1];
    if (net != cur) {
      if (cur >= 0) {
        atomicAdd(&sxp[cur], axp); atomicAdd(&sxn[cur], axn);
        atomicAdd(&syp[cur], ayp); atomicAdd(&syn[cur], ayn);
      }
      cur = net;
      mxp = key2f(kxp[net]); mxn = key2f(kxn[net]);
      myp = key2f(kyp[net]); myn = key2f(kyn[net]);
      axp = axn = ayp = ayn = 0.f;
    }
    axp += __expf(( x - mxp) * INV_GAMMA);
    axn += __expf((-x - mxn) * INV_GAMMA);
    ayp += __expf(( y - myp) * INV_GAMMA);
    ayn += __expf((-y - myn) * INV_GAMMA);
  }
  if (cur >= 0) {
    atomicAdd(&sxp[cur], axp); atomicAdd(&sxn[cur], axn);
    atomicAdd(&syp[cur], ayp); atomicAdd(&syn[cur], ayn);
  }
}

// ---------- pass 3: total = sum over nets/components of m + gamma*log(max(s,1e-30))
__global__ void __launch_bounds__(BLOCK) finalize(
    const unsigned* __restrict__ keys, const float* __restrict__ sums,
    float* __restrict__ out) {
  float acc = 0.f;
  for (int n = blockIdx.x * BLOCK + threadIdx.x; n < NUM_NETS;
       n += gridDim.x * BLOCK) {
#pragma unroll
    for (int c = 0; c < 4; ++c) {
      const unsigned k = keys[(size_t)c * NUM_NETS + n];
      if (k != 0u) {   // key==0 -> empty net -> contributes 0
        const float s = sums[(size_t)c * NUM_NETS + n];
        acc += key2f(k) + GAMMA_F * __logf(fmaxf(s, 1e-30f));
      }
    }
  }
  __shared__ float red[BLOCK];
  red[threadIdx.x] = acc;
  __syncthreads();
  for (int off = BLOCK / 2; off > 0; off >>= 1) {
    if (threadIdx.x < off) red[threadIdx.x] += red[threadIdx.x + off];
    __syncthreads();
  }
  if (threadIdx.x == 0) atomicAdd(out, red[0]);
}

extern "C" void kernel_launch(void* const* d_in, const int* in_sizes, int n_in,
                              void* d_out, int out_size, void* d_ws, size_t ws_size,
                              hipStream_t stream) {
  const float* cells_pos   = (const float*)d_in[0];   // [NUM_CELLS,2]
  const float* pin_offsets = (const float*)d_in[1];   // [NUM_PINS,2]
  const int*   pin_cell    = (const int*)d_in[2];     // [NUM_PINS]
  const int*   pin_net     = (const int*)d_in[3];     // [NUM_PINS], sorted

  unsigned* keys = (unsigned*)d_ws;                                    // 4*N uints
  float*    sums = (float*)((char*)d_ws + (size_t)4 * NUM_NETS * 4);   // 4*N floats

  hipMemsetAsync(d_ws, 0, (size_t)8 * NUM_NETS * 4, stream);  // keys=0 (empty), sums=0
  hipMemsetAsync(d_out, 0, sizeof(float), stream);

  const int nblk = (NUM_PINS + TILE - 1) / TILE;
  pass_minmax<<<nblk, BLOCK, 0, stream>>>(cells_pos, pin_offsets, pin_cell,
                                          pin_net, keys);
  pass_sums<<<nblk, BLOCK, 0, stream>>>(cells_pos, pin_offsets, pin_cell,
                                        pin_net, keys, sums);
  finalize<<<2048, BLOCK, 0, stream>>>(keys, sums, (float*)d_out);
}